// KTM_22110491640579
// MI455X (gfx1250) — compile-verified
//
#include <hip/hip_runtime.h>
#include <math.h>

// Problem constants (match reference)
#define F_NUM  2048
#define EMB_D  64
#define BATCH  64
#define KSPLIT 8
#define KCHUNK (F_NUM / KSPLIT)   // 256

typedef __attribute__((ext_vector_type(2))) float v2f;
typedef __attribute__((ext_vector_type(8))) float v8f;

// Kernel 1: split-K GEMM  s_part[c] += data[:, cK:(c+1)K] @ embed[cK:(c+1)K, :]
// Grid: KSPLIT blocks x 512 threads (16 waves). Wave w owns output tile
// (mt = w/4, nt = w%4) of the 64x64 result, accumulating over its K chunk
// with V_WMMA_F32_16X16X4_F32 (full-precision f32 matrix pipe).
__global__ __launch_bounds__(512) void fm_gemm_splitk(
    const float* __restrict__ data,    // (BATCH, F_NUM) row-major
    const float* __restrict__ embed,   // (F_NUM, EMB_D) row-major
    float* __restrict__ s_part)        // (KSPLIT, BATCH, EMB_D)
{
  const int wave = threadIdx.x >> 5;     // 0..15
  const int lane = threadIdx.x & 31;     // wave32
  const int mt = (wave >> 2) * 16;       // row-tile base (batch dim)
  const int nt = (wave & 3) * 16;        // col-tile base (embed dim)
  const int kbase = blockIdx.x * KCHUNK;

  // A fragment (16x4, MxK): lanes 0-15 carry K={0,1}, lanes 16-31 K={2,3};
  // row M = lane&15. Each lane reads a contiguous float pair of `data`.
  const int mrow  = mt + (lane & 15);
  const int khalf = (lane >> 4) * 2;     // 0 or 2
  // B fragment (4x16, KxN): VGPR0 -> K=khalf, VGPR1 -> K=khalf+1; N = lane&15.
  const int ncol  = nt + (lane & 15);

  const float* __restrict__ arow = data  + (size_t)mrow * F_NUM + kbase + khalf;
  const float* __restrict__ bcol = embed + (size_t)(kbase + khalf) * EMB_D + ncol;

  v8f acc = {};
  #pragma unroll 4
  for (int k = 0; k < KCHUNK; k += 4) {
    v2f a, b;
    a.x = arow[k + 0];
    a.y = arow[k + 1];
    b.x = bcol[(k + 0) * EMB_D];
    b.y = bcol[(k + 1) * EMB_D];
    // 8 args: (neg_a, A, neg_b, B, c_mod, C, reuse_a, reuse_b)
    acc = __builtin_amdgcn_wmma_f32_16x16x4_f32(
        false, a, false, b, (short)0, acc, false, false);
  }

  // C/D layout: VGPR v holds row v (lanes 0-15) / row v+8 (lanes 16-31),
  // col = lane&15.
  const int crow = (lane >> 4) * 8;
  const int ccol = lane & 15;
  float* __restrict__ out =
      s_part + ((size_t)blockIdx.x * BATCH + mt + crow) * EMB_D + nt + ccol;
  #pragma unroll
  for (int v = 0; v < 8; ++v)
    out[(size_t)v * EMB_D] = acc[v];
}

// Kernel 2: per-batch reduction. One block per batch row b:
//   lin  = sum_f data[b,f]*bias[f]        (coalesced over f)
//   s[d] = sum_c s_part[c,b,d]  (fixed order -> deterministic); dot = sum_d s[d]^2
//   out[b] = sigmoid(gb + lin + dot)
__global__ __launch_bounds__(256) void fm_finalize(
    const float* __restrict__ data,
    const float* __restrict__ bias,     // (F_NUM, 1)
    const float* __restrict__ gbias,    // (1, 1)
    const float* __restrict__ s_part,   // (KSPLIT, BATCH, EMB_D)
    float* __restrict__ out)            // (BATCH,)
{
  __shared__ float red[256];
  const int b   = blockIdx.x;
  const int tid = threadIdx.x;

  float p = 0.0f;
  for (int f = tid; f < F_NUM; f += 256)
    p += data[(size_t)b * F_NUM + f] * bias[f];

  if (tid < EMB_D) {
    float sv = 0.0f;
    #pragma unroll
    for (int c = 0; c < KSPLIT; ++c)
      sv += s_part[((size_t)c * BATCH + b) * EMB_D + tid];
    p += sv * sv;
  }

  red[tid] = p;
  __syncthreads();
  #pragma unroll
  for (int step = 128; step > 0; step >>= 1) {
    if (tid < step) red[tid] += red[tid + step];
    __syncthreads();
  }

  if (tid == 0) {
    float z = gbias[0] + red[0];
    out[b] = 1.0f / (1.0f + __expf(-z));
  }
}

extern "C" void kernel_launch(void* const* d_in, const int* in_sizes, int n_in,
                              void* d_out, int out_size, void* d_ws, size_t ws_size,
                              hipStream_t stream) {
  (void)in_sizes; (void)n_in; (void)out_size; (void)ws_size;

  const float* data  = (const float*)d_in[0];  // (64, 2048) f32
  const float* embed = (const float*)d_in[1];  // (2048, 64) f32
  const float* bias  = (const float*)d_in[2];  // (2048, 1)  f32
  const float* gb    = (const float*)d_in[3];  // (1, 1)     f32
  float* out    = (float*)d_out;               // (64,)      f32
  float* s_part = (float*)d_ws;                // KSPLIT*64*64 f32 = 128 KB

  fm_gemm_splitk<<<dim3(KSPLIT), dim3(512), 0, stream>>>(data, embed, s_part);
  fm_finalize<<<dim3(BATCH), dim3(256), 0, stream>>>(data, bias, gb, s_part, out);
}